// TimeDependentRatchetPolicyBW_249108103278
// MI455X (gfx1250) — compile-verified
//
#include <hip/hip_runtime.h>
#include <stdint.h>

typedef float v2f __attribute__((ext_vector_type(2)));
typedef float v8f __attribute__((ext_vector_type(8)));

// Constants from the reference (folded in double, emitted as f32 like JAX does)
#define K_NEGB  0.07f          // -(RHO - R - 0.5*THETA_EFF^2)
#define K_2AQ   0.16f          // 2*A,  A = 0.5*THETA_EFF^2 = 0.08
#define K_4AQ   0.32f          // 4*A
#define K_RHO   0.03f
#define K_R     0.02f
#define K_GAMMA 3.0f

__global__ __launch_bounds__(128) void ratchet_policy_wmma_kernel(
    const float* __restrict__ X,      // (n, 2) wealth, habit
    const float* __restrict__ TmT,    // (n, 1) tau
    const float* __restrict__ u_star, // (16,)
    float* __restrict__ out,          // (n, 17), 136 MB -> 32-bit offsets suffice
    int n)
{
    const int lane = threadIdx.x & 31;             // wave32
    const int wave = (blockIdx.x * blockDim.x + threadIdx.x) >> 5;
    const int base = wave * 32;                    // 32 rows per wave
    if (base >= n) return;                         // wave-uniform exit only

    // Clamp (not predicate) the tail so EXEC stays all-ones for WMMA.
    const int row  = base + lane;
    const uint32_t rowc = (uint32_t)(row < n ? row : (n - 1));

    // ---- contiguous loads: 256B X, 128B TmT per wave ----
    const float wealth = X[rowc * 2u + 0u];
    const float habit  = X[rowc * 2u + 1u];
    const float tau    = TmT[rowc];

    // ---- per-row scalar policy math (reference order) ----
    const float tau_c   = fmaxf(tau, 0.0f);
    const float Cq      = -K_RHO * (1.0f - expf(-K_RHO * tau_c));
    const float disc    = fmaxf(K_NEGB * K_NEGB - K_4AQ * Cq, 0.0f);
    const float n_minus = (K_NEGB - sqrtf(disc)) / K_2AQ;
    const float rs      = 1.0f / (1.0f - n_minus);
    const float rstar   = (tau_c <= 0.0f) ? 1.0f : rs;
    const float ann     = (1.0f - expf(-K_R * tau)) / K_R;   // raw tau, per reference
    const float gm      = fmaxf(K_GAMMA - rstar, 1e-6f);
    const float X_th    = (K_GAMMA / gm) * habit * ann;
    const float c_on_b  = wealth * (gm / K_GAMMA) / fmaxf(ann, 1e-10f);
    float C = (wealth > X_th) ? fmaxf(c_on_b, habit) : habit; // ALPHA_RAISE == 1
    C = fminf(C, fmaxf(habit, 0.5f * wealth));
    const float pvC = ann * C;
    float dw = (wealth - pvC) / fmaxf(wealth, 1e-12f);
    dw = fminf(fmaxf(dw, 0.0f), 1.0f);

    // ---- outer product u = dw ⊗ u_star via V_WMMA_F32_16X16X4_F32 ----
    // B (4x16): lane holds column N = lane&15; VGPR0 = K=0 row = u_star[N], VGPR1 = 0.
    v2f bmat;
    bmat[0] = u_star[lane & 15];
    bmat[1] = 0.0f;

    // A (16x4): lanes 0-15 hold K=0 (VGPR0) / K=1 (VGPR1); lanes 16-31 hold K=2/K=3.
    // Only K=0 nonzero -> exact single-product outer product.
    const float dwHi = __shfl_down(dw, 16, 32);    // lanes 0-15 receive rows base+16..31
    v2f a0; a0[0] = (lane < 16) ? dw   : 0.0f; a0[1] = 0.0f;
    v2f a1; a1[0] = (lane < 16) ? dwHi : 0.0f; a1[1] = 0.0f;

    const v8f czero = {};
    v8f d0 = __builtin_amdgcn_wmma_f32_16x16x4_f32(
        false, a0, false, bmat, (short)0, czero, false, false);   // rows base+0..15
    v8f d1 = __builtin_amdgcn_wmma_f32_16x16x4_f32(
        false, a1, false, bmat, (short)0, czero, false, false);   // rows base+16..31

    // ---- scatter D per C/D layout: VGPR v, lane L -> row v + (L>=16)*8, col L&15 ----
    // 32-bit element offsets keep stores in saddr+voffset form (no 64-bit mads).
    const uint32_t col   = (uint32_t)(lane & 15);
    const int      mOff  = (lane >> 4) << 3;       // 0 or 8
    const int      nm1   = n - 1;
#pragma unroll
    for (int v = 0; v < 8; ++v) {
        int r0 = base + v + mOff;
        int r1 = base + 16 + v + mOff;
        r0 = r0 < n ? r0 : nm1;
        r1 = r1 < n ? r1 : nm1;
        out[(uint32_t)r0 * 17u + col] = d0[v];
        out[(uint32_t)r1 * 17u + col] = d1[v];
    }

    // ---- consumption column ----
    out[rowc * 17u + 16u] = C;
}

extern "C" void kernel_launch(void* const* d_in, const int* in_sizes, int n_in,
                              void* d_out, int out_size, void* d_ws, size_t ws_size,
                              hipStream_t stream)
{
    const float* X      = (const float*)d_in[0];
    const float* TmT    = (const float*)d_in[1];
    const float* u_star = (const float*)d_in[2];
    float* out          = (float*)d_out;

    const int n      = in_sizes[1];                // BATCH (TmT element count)
    const int waves  = (n + 31) / 32;              // 32 rows per wave32
    const int blocks = (waves + 3) / 4;            // 4 waves per 128-thread block

    hipLaunchKernelGGL(ratchet_policy_wmma_kernel,
                       dim3(blocks), dim3(128), 0, stream,
                       X, TmT, u_star, out, n);
}